// ChamferLossLayer_481036337470
// MI455X (gfx1250) — compile-verified
//
#include <hip/hip_runtime.h>

typedef __attribute__((ext_vector_type(2))) float v2f;
typedef __attribute__((ext_vector_type(8))) float v8f;

#define NPTS   8192
#define NBATCH 4
#define STRIP  128      // rows of query cloud per block (8 waves x 16)
#define CHUNK  2048     // reference points staged in LDS per iteration
#define THREADS 256

// One direction of chamfer: for each query point p in X, min over all q in Y of
// (-2 x.y + ||y||^2), summed over the block's 128 rows -> part[batch*64 + blk].
// The ||x||^2 term is added later as a global mean (min over q commutes with it).
__launch_bounds__(THREADS)
__global__ void chamfer_dir_kernel(const float* __restrict__ X,
                                   const float* __restrict__ Y,
                                   float* __restrict__ part) {
  __shared__ float4 sB[CHUNK];   // [-2bx, -2by, -2bz, ||b||^2]
  __shared__ float wpart[8];

  const int n    = blockIdx.y;
  const int tid  = threadIdx.x;
  const int lane = tid & 31;
  const int wave = tid >> 5;

  const float* __restrict__ Xn = X + (size_t)n * NPTS * 3;
  const float* __restrict__ Yn = Y + (size_t)n * NPTS * 3;

  // A' tile for this wave: row M = [x, y, z, 1].
  // 32-bit 16x4 A layout: lanes 0-15 hold (K0,K1), lanes 16-31 hold (K2,K3), M = lane%16.
  const int row = blockIdx.x * STRIP + wave * 16 + (lane & 15);
  v2f a;
  {
    const float ax = Xn[row * 3 + 0];
    const float ay = Xn[row * 3 + 1];
    const float az = Xn[row * 3 + 2];
    if (lane < 16) { a.x = ax; a.y = ay; }
    else           { a.x = az; a.y = 1.0f; }
  }

  v8f run;
#pragma unroll
  for (int i = 0; i < 8; ++i) run[i] = __builtin_inff();

  for (int cb = 0; cb < NPTS; cb += CHUNK) {
    // Cooperative stage: precompute B' into LDS.
    for (int i = tid; i < CHUNK; i += THREADS) {
      const int q = cb + i;
      const float bx = Yn[q * 3 + 0];
      const float by = Yn[q * 3 + 1];
      const float bz = Yn[q * 3 + 2];
      sB[i] = make_float4(-2.0f * bx, -2.0f * by, -2.0f * bz,
                          bx * bx + by * by + bz * bz);
    }
    __syncthreads();

    const float2* __restrict__ sB2 = reinterpret_cast<const float2*>(sB);
#pragma unroll 4
    for (int t = 0; t < CHUNK / 16; ++t) {
      // 32-bit 4x16 B layout: lanes 0-15 hold (K0,K1), lanes 16-31 hold (K2,K3), N = lane%16.
      const float2 bv = sB2[(t * 16 + (lane & 15)) * 2 + (lane >> 4)];
      v2f b; b.x = bv.x; b.y = bv.y;
      v8f c = {};
      c = __builtin_amdgcn_wmma_f32_16x16x4_f32(false, a, false, b,
                                                (short)0, c, false, false);
#pragma unroll
      for (int i = 0; i < 8; ++i) run[i] = fminf(run[i], c[i]);
    }
    __syncthreads();
  }

  // C layout: VGPR v, lanes 0-15 -> (M=v, N=lane), lanes 16-31 -> (M=v+8, N=lane-16).
  // Row-min = min across the 16 lanes of each half (xor masks 1,2,4,8 stay in-half).
  float s = 0.0f;
#pragma unroll
  for (int v = 0; v < 8; ++v) {
    float m = run[v];
    m = fminf(m, __shfl_xor(m, 1, 32));
    m = fminf(m, __shfl_xor(m, 2, 32));
    m = fminf(m, __shfl_xor(m, 4, 32));
    m = fminf(m, __shfl_xor(m, 8, 32));
    s += m;                       // per-half sum of its 8 row-mins
  }
  s += __shfl_xor(s, 16, 32);     // combine halves -> sum over 16 rows

  if (lane == 0) wpart[wave] = s;
  __syncthreads();
  if (tid == 0) {
    float acc = 0.0f;
#pragma unroll
    for (int w = 0; w < 8; ++w) acc += wpart[w];
    part[n * (NPTS / STRIP) + blockIdx.x] = acc;
  }
}

// Per batch: out[n] = (sum_min_dir0 + sum ||c1||^2)/8192 + (sum_min_dir1 + sum ||c2||^2)/8192
__launch_bounds__(THREADS)
__global__ void chamfer_finalize_kernel(const float* __restrict__ c1,
                                        const float* __restrict__ c2,
                                        const float* __restrict__ part,
                                        float* __restrict__ out) {
  __shared__ float red[THREADS];
  const int n = blockIdx.x;
  const int tid = threadIdx.x;
  const int nblk = NPTS / STRIP;  // 64

  float s = 0.0f;
  if (tid < nblk) {
    s += part[n * nblk + tid];                       // dir 0
    s += part[NBATCH * nblk + n * nblk + tid];       // dir 1
  }

  const float* __restrict__ p1 = c1 + (size_t)n * NPTS * 3;
  const float* __restrict__ p2 = c2 + (size_t)n * NPTS * 3;
  for (int p = tid; p < NPTS; p += THREADS) {
    float x = p1[p * 3 + 0], y = p1[p * 3 + 1], z = p1[p * 3 + 2];
    s += x * x + y * y + z * z;
    x = p2[p * 3 + 0]; y = p2[p * 3 + 1]; z = p2[p * 3 + 2];
    s += x * x + y * y + z * z;
  }

  red[tid] = s;
  __syncthreads();
  for (int off = THREADS / 2; off > 0; off >>= 1) {
    if (tid < off) red[tid] += red[tid + off];
    __syncthreads();
  }
  if (tid == 0) out[n] = red[0] * (1.0f / (float)NPTS);
}

extern "C" void kernel_launch(void* const* d_in, const int* in_sizes, int n_in,
                              void* d_out, int out_size, void* d_ws, size_t ws_size,
                              hipStream_t stream) {
  const float* c1 = (const float*)d_in[0];
  const float* c2 = (const float*)d_in[1];
  float* out  = (float*)d_out;
  float* part = (float*)d_ws;   // 2 dirs * 4 batches * 64 blocks = 512 floats

  dim3 grid(NPTS / STRIP, NBATCH);
  // Direction 0: queries = cloud1, references = cloud2
  chamfer_dir_kernel<<<grid, THREADS, 0, stream>>>(c1, c2, part);
  // Direction 1: queries = cloud2, references = cloud1
  chamfer_dir_kernel<<<grid, THREADS, 0, stream>>>(c2, c1, part + NBATCH * (NPTS / STRIP));

  chamfer_finalize_kernel<<<NBATCH, THREADS, 0, stream>>>(c1, c2, part, out);
}